// SwapAlign2Nat_35304631173565
// MI455X (gfx1250) — compile-verified
//
#include <hip/hip_runtime.h>
#include <cstdint>

// SwapAlign2Nat (lam=2, pad=-6) for X:(8,1024,64,64)f32 -> Y:(8,256,128,128)f32.
// Memory-bound: 268MB traffic -> ~11.5us floor @ 23.3TB/s. Strategy:
//   * one workgroup per (n, vout, uout): average the 2x2 channel block into a
//     padded 80x80 LDS tile (pad = -6 halo kills all bounds checks),
//   * stage the 4 source planes via gfx1250 async global->LDS copies (ASYNCcnt),
//   * each thread produces 2x2 output quads (shared floor trick: both y parities
//     share yi=qy+v-8), stores via non-temporal v2f to keep L2 for the reads.

typedef float v2f __attribute__((ext_vector_type(2)));
typedef float v4f __attribute__((ext_vector_type(4)));
typedef int   v4i __attribute__((ext_vector_type(4)));

#define PAD_VAL (-6.0f)
#define STR 80

#if defined(__has_builtin)
#  if __has_builtin(__builtin_amdgcn_global_load_async_to_lds_b128)
#    define HAVE_ASYNC 1
#  endif
#endif
#ifndef HAVE_ASYNC
#  define HAVE_ASYNC 0
#endif

typedef __attribute__((address_space(1))) v4i* gp4;   // global int4*
typedef __attribute__((address_space(3))) v4i* lp4;   // LDS int4*

__device__ __forceinline__ void cp16(float* l, const float* g) {
#if HAVE_ASYNC
  // generic global ptr and AS(1) ptr share the same 64-bit value (inttoptr);
  // ISA 10.2: generic LDS address low 32 bits == workgroup-relative LDS offset
  __builtin_amdgcn_global_load_async_to_lds_b128(
      (gp4)(uintptr_t)g, (lp4)(unsigned int)(uintptr_t)l, 0, 0);
#else
  *(v4f*)l = *(const v4f*)g;
#endif
}

__device__ __forceinline__ void wait_cp() {
#if HAVE_ASYNC
#  if __has_builtin(__builtin_amdgcn_s_wait_asynccnt)
  __builtin_amdgcn_s_wait_asynccnt(0);
#  else
  asm volatile("s_wait_asynccnt 0" ::: "memory");
#  endif
#endif
}

__global__ __launch_bounds__(256, 1) void sa2n_kernel(const float* __restrict__ X,
                                                      float* __restrict__ Y) {
  __shared__ float raw[8192];        // two 64x64 planes, 32 KiB
  __shared__ float Ap[STR * STR];    // padded averaged tile, 25.6 KiB

  const int tid = threadIdx.x;
  const int b   = blockIdx.x;        // 0..2047
  const int n   = b >> 8;
  const int vo  = (b >> 4) & 15;
  const int uo  = b & 15;

  // channel index c = vi*32 + ui; planes are contiguous 64*64 = 4096 floats
  const size_t cbase = (size_t)n * 1024 + (size_t)(2 * vo) * 32 + (size_t)(2 * uo);
  const float* p00 = X + (cbase << 12);          // (2v,   2u)
  const float* p01 = X + ((cbase + 1) << 12);    // (2v,   2u+1)
  const float* p10 = X + ((cbase + 32) << 12);   // (2v+1, 2u)
  const float* p11 = X + ((cbase + 33) << 12);   // (2v+1, 2u+1)

  // ---- pass 1: stage planes (2v,2u) and (2v,2u+1) asynchronously ----
#pragma unroll
  for (int k = 0; k < 4; ++k) {
    const int q = (k << 8) + tid;                // float4 index 0..1023
    cp16(&raw[q << 2],          p00 + (q << 2));
    cp16(&raw[4096 + (q << 2)], p01 + (q << 2));
  }
  // fill the -6 halo (entire tile) while the async copies are in flight
  for (int i = tid; i < STR * STR; i += 256) Ap[i] = PAD_VAL;
  wait_cp();
  __syncthreads();

  // interior partial sum (overwrites the PAD fill in [8..71]^2)
#pragma unroll
  for (int k = 0; k < 4; ++k) {
    const int q = (k << 8) + tid;
    const int h = q >> 4;
    const int w = (q & 15) << 2;
    v4f a = *(const v4f*)&raw[q << 2];
    v4f c = *(const v4f*)&raw[4096 + (q << 2)];
    *(v4f*)&Ap[(h + 8) * STR + (w + 8)] = a + c;
  }
  __syncthreads();

  // ---- pass 2: stage planes (2v+1,2u) and (2v+1,2u+1) ----
#pragma unroll
  for (int k = 0; k < 4; ++k) {
    const int q = (k << 8) + tid;
    cp16(&raw[q << 2],          p10 + (q << 2));
    cp16(&raw[4096 + (q << 2)], p11 + (q << 2));
  }
  wait_cp();
  __syncthreads();

#pragma unroll
  for (int k = 0; k < 4; ++k) {
    const int q = (k << 8) + tid;
    const int h = q >> 4;
    const int w = (q & 15) << 2;
    v4f a = *(const v4f*)&raw[q << 2];
    v4f c = *(const v4f*)&raw[4096 + (q << 2)];
    float* ap = &Ap[(h + 8) * STR + (w + 8)];
    v4f s = *(v4f*)ap;
    *(v4f*)ap = 0.25f * (s + a + c);             // A = mean of the 4 channels
  }
  __syncthreads();

  // ---- bilinear 2x upsample with per-(v,u) shift; 2x2 quad per thread ----
  float* out = Y + ((size_t)((n << 8) + (vo << 4) + uo) << 14);   // *16384
#pragma unroll 4
  for (int k = 0; k < 16; ++k) {
    const int Q    = (k << 8) + tid;             // quad id 0..4095
    const int qy   = Q >> 6;
    const int qx   = Q & 63;
    const int base = (qy + vo) * STR + (qx + uo);  // padded row = yi+8 = qy+vo
    const float a00 = Ap[base];
    const float a01 = Ap[base + 1];
    const float a10 = Ap[base + STR];
    const float a11 = Ap[base + STR + 1];
    // x-lerps for even/odd output columns, rows yi and yi+1
    const float e0 = 0.75f * a00 + 0.25f * a01;
    const float o0 = 0.25f * a00 + 0.75f * a01;
    const float e1 = 0.75f * a10 + 0.25f * a11;
    const float o1 = 0.25f * a10 + 0.75f * a11;
    v2f r0, r1;
    r0.x = 0.75f * e0 + 0.25f * e1;  r0.y = 0.75f * o0 + 0.25f * o1;  // y even
    r1.x = 0.25f * e0 + 0.75f * e1;  r1.y = 0.25f * o0 + 0.75f * o1;  // y odd
    float* o = out + (qy << 8) + (qx << 1);      // (2qy)*128 + 2qx
    __builtin_nontemporal_store(r0, (v2f*)o);
    __builtin_nontemporal_store(r1, (v2f*)(o + 128));
  }
}

extern "C" void kernel_launch(void* const* d_in, const int* in_sizes, int n_in,
                              void* d_out, int out_size, void* d_ws, size_t ws_size,
                              hipStream_t stream) {
  const float* X = (const float*)d_in[0];
  float* Y = (float*)d_out;
  (void)in_sizes; (void)n_in; (void)out_size; (void)d_ws; (void)ws_size;
  sa2n_kernel<<<dim3(2048), dim3(256), 0, stream>>>(X, Y);
}